// CGPooling_9680856285727
// MI455X (gfx1250) — compile-verified
//
#include <hip/hip_runtime.h>

// Segment mean (sorted segment_ids) for MI455X / gfx1250.
//
// Roofline: traffic = 1 GiB read of atom_features (+4 MB out) -> ~46 us at
// 23.3 TB/s HBM; compute is negligible. Strategy: one workgroup per segment
// (ids are sorted => contiguous row range found by binary search), stream rows
// once with non-temporal b128 loads, accumulate column sums on the matrix pipe
// via V_WMMA_F32_16X16X4_F32 with an all-ones A matrix (D[m][n] = sum_k B[k][n]
// + C[m][n]), divide by count, store. No atomics, no workspace, deterministic.

typedef __attribute__((ext_vector_type(2))) float v2f;
typedef __attribute__((ext_vector_type(4))) float v4f;
typedef __attribute__((ext_vector_type(8))) float v8f;

__device__ __forceinline__ long lower_bound_i32(const int* __restrict__ ids,
                                                long n, int key) {
    long lo = 0, hi = n;
    while (lo < hi) {
        long mid = (lo + hi) >> 1;
        if (ids[mid] < key) lo = mid + 1; else hi = mid;
    }
    return lo;
}

__global__ __launch_bounds__(64) void seg_mean_wmma(
    const float* __restrict__ A,   // [n_rows, 128]
    const int*   __restrict__ ids, // [n_rows], sorted
    float*       __restrict__ out, // [num_segments, 128]
    long n_rows) {
    const int seg = blockIdx.x;

    // Contiguous row range of this segment (uniform across the block).
    const long start = lower_bound_i32(ids, n_rows, seg);
    const long end   = lower_bound_i32(ids, n_rows, seg + 1);

    const int lane = threadIdx.x & 31;
    const int wv   = threadIdx.x >> 5;   // 0..1 : which 64-feature half
    const int grp  = lane >> 4;          // 0..1 : which row pair of the 4-row group
    const int j    = lane & 15;          // WMMA column index
    const int col  = wv * 64 + 4 * j;    // feature base (4 consecutive features)

    // 4 accumulators: feature col+0..col+3 mapped to WMMA column n = lane&15.
    v8f c0 = {}, c1 = {}, c2 = {}, c3 = {};
    const v2f ones = {1.0f, 1.0f};       // all-ones 16x4 A matrix (layout-proof)

    long i = start;
    const long mainEnd = start + ((end - start) & ~3L);

    for (; i < mainEnd; i += 4) {
        const long r0 = i + 2 * grp;                 // lanes 0-15: rows i,i+1 ; 16-31: i+2,i+3
        const float* p0 = A + r0 * 128 + col;
        v4f L0 = __builtin_nontemporal_load((const v4f*)p0);          // row r0, 4 features
        v4f L1 = __builtin_nontemporal_load((const v4f*)(p0 + 128));  // row r0+1
        __builtin_prefetch(p0 + 32 * 128, 0, 1);     // speculative stream prefetch

        v2f b0 = {L0[0], L1[0]};
        v2f b1 = {L0[1], L1[1]};
        v2f b2 = {L0[2], L1[2]};
        v2f b3 = {L0[3], L1[3]};
        c0 = __builtin_amdgcn_wmma_f32_16x16x4_f32(false, ones, false, b0, (short)0, c0, false, false);
        c1 = __builtin_amdgcn_wmma_f32_16x16x4_f32(false, ones, false, b1, (short)0, c1, false, false);
        c2 = __builtin_amdgcn_wmma_f32_16x16x4_f32(false, ones, false, b2, (short)0, c2, false, false);
        c3 = __builtin_amdgcn_wmma_f32_16x16x4_f32(false, ones, false, b3, (short)0, c3, false, false);
    }

    // Tail: <4 remaining rows, predicated loads (zeros contribute nothing).
    if (i < end) {
        const long r0 = i + 2 * grp;
        const long r1 = r0 + 1;
        v4f L0 = {}, L1 = {};
        if (r0 < end) L0 = *(const v4f*)(A + r0 * 128 + col);
        if (r1 < end) L1 = *(const v4f*)(A + r1 * 128 + col);
        v2f b0 = {L0[0], L1[0]};
        v2f b1 = {L0[1], L1[1]};
        v2f b2 = {L0[2], L1[2]};
        v2f b3 = {L0[3], L1[3]};
        c0 = __builtin_amdgcn_wmma_f32_16x16x4_f32(false, ones, false, b0, (short)0, c0, false, false);
        c1 = __builtin_amdgcn_wmma_f32_16x16x4_f32(false, ones, false, b1, (short)0, c1, false, false);
        c2 = __builtin_amdgcn_wmma_f32_16x16x4_f32(false, ones, false, b2, (short)0, c2, false, false);
        c3 = __builtin_amdgcn_wmma_f32_16x16x4_f32(false, ones, false, b3, (short)0, c3, false, false);
    }

    // With all-ones A, every element of C in lane L equals the column sum for
    // n = L&15, so c*[0] is valid in every lane; lanes 0-15 of each wave write
    // the 64 features owned by that wave (2 waves cover all 128).
    const long cnt = end - start;
    const float inv = 1.0f / (float)(cnt > 0 ? cnt : 1);
    if (grp == 0) {
        v4f r = { c0[0] * inv, c1[0] * inv, c2[0] * inv, c3[0] * inv };
        *(v4f*)(out + (long)seg * 128 + col) = r;
    }
}

extern "C" void kernel_launch(void* const* d_in, const int* in_sizes, int n_in,
                              void* d_out, int out_size, void* d_ws, size_t ws_size,
                              hipStream_t stream) {
    const float* A   = (const float*)d_in[0];  // atom_features [N,128] f32
    const int*   ids = (const int*)d_in[1];    // segment_ids   [N] (sorted)
    float*       out = (float*)d_out;          // [B,128] f32

    const long n_rows   = (long)in_sizes[1];   // N
    const int  num_segs = out_size / 128;      // B

    seg_mean_wmma<<<num_segs, 64, 0, stream>>>(A, ids, out, n_rows);
}